// RPNPostProcessor_21809843929142
// MI455X (gfx1250) — compile-verified
//
#include <hip/hip_runtime.h>
#include <hip/hip_bf16.h>
#include <stdint.h>

typedef unsigned int u32;
typedef unsigned long long u64;
typedef __attribute__((ext_vector_type(4))) u32 v4u;
typedef __attribute__((ext_vector_type(8))) u32 v8u;

#define BATCH     4
#define A_NUM     2
#define HT_TOT    1048576   /* B*262144 */
#define LOCS      262144
#define NPB       524288    /* LOCS*A per batch */
#define TOTAL     2097152   /* A*HT_TOT */
#define PRE_NMS   2048
#define POST_NMS  512
#define NMS_THR   0.7f

/* ---- workspace layout (u32 units) ---- */
#define OFF_HIST1 0
#define OFF_HIST2 (OFF_HIST1 + BATCH*2048)
#define OFF_HIST3 (OFF_HIST2 + BATCH*2048)
#define OFF_SEL   (OFF_HIST3 + BATCH*1024)     /* 8 u32/batch: bin1,need1,bin2,need2,kth,cg,eq,pad */
#define OFF_CNT   (OFF_SEL  + BATCH*8)         /* cntG,cntE per batch */
#define ZERO_U32  (OFF_CNT  + BATCH*2)
#define OFF_TKEY  ZERO_U32
#define OFF_TIDX  (OFF_TKEY + BATCH*PRE_NMS)
#define OFF_SCORE (OFF_TIDX + BATCH*PRE_NMS)
#define OFF_BOX   (OFF_SCORE+ BATCH*PRE_NMS)       /* B*2048*7 f32 */
#define OFF_AUX   (OFF_BOX  + BATCH*PRE_NMS*7)     /* x1,x2,y1,y2,z1,z2,vol */
#define OFF_SUP   (OFF_AUX  + BATCH*PRE_NMS*7)     /* B*2048*64 u32 bitmask */

__device__ __forceinline__ u32 fkey(float f) {
  u32 u = __float_as_uint(f);
  return (u & 0x80000000u) ? ~u : (u | 0x80000000u);
}
__device__ __forceinline__ float kinv(u32 k) {
  u32 u = (k & 0x80000000u) ? (k & 0x7fffffffu) : ~k;
  return __uint_as_float(u);
}

/* ================= Pass 1 histogram: TDM double-buffered objectness -> LDS ================= */
#define H1_CHUNK 4096   /* floats per TDM tile = 16 KB */
#define H1_CPB   4      /* chunks per block; 512 chunks / 4 = 128 blocks */

/* Issue one Tensor-Data-Mover DMA: 1-row tile of H1_CHUNK fp32 elements, global -> LDS.
   D# packing per CDNA5 ISA §8.3/8.4: group0 = {count=1, lds_addr, global_addr, type=2},
   group1 = {data_size=4B, tensor_dim0=tile_dim0=stride=H1_CHUNK, tensor_dim1=tile_dim1=1}. */
__device__ __forceinline__ void tdm_issue(const void* gptr, void* lptr) {
  u64 ga = (u64)(uintptr_t)gptr;
  u32 lds_off = (u32)(u64)(uintptr_t)lptr;    /* flat LDS addr: low 32 bits = offset */
  v4u g0; v8u g1;
  g0.x = 1u;                                           /* count=1, user descriptor */
  g0.y = lds_off;                                      /* lds_addr @ [63:32] */
  g0.z = (u32)ga;                                      /* global_addr lo */
  g0.w = (u32)((ga >> 32) & 0x01FFFFFFu) | (2u << 30); /* global_addr hi, type=2 */
  g1.s0 = (2u << 16);                                  /* data_size=4B, no multicast */
  g1.s1 = ((u32)H1_CHUNK & 0xFFFFu) << 16;             /* tensor_dim0 lo16 @ bit48 */
  g1.s2 = (((u32)H1_CHUNK >> 16) & 0xFFFFu)            /* tensor_dim0 hi16 */
        | (1u << 16);                                  /* tensor_dim1 lo16 = 1 */
  g1.s3 = ((u32)H1_CHUNK) << 16;                       /* tile_dim0 @ [127:112] */
  g1.s4 = 1u;                                          /* tile_dim1=1, tile_dim2=0 */
  g1.s5 = (u32)H1_CHUNK;                               /* tensor_dim0_stride lo32 */
  g1.s6 = 0u;
  g1.s7 = 0u;
  asm volatile("tensor_load_to_lds %0, %1" :: "s"(g0), "s"(g1) : "memory");
}

__global__ __launch_bounds__(256) void k_hist1(const float* __restrict__ obj, u32* __restrict__ ws) {
  __shared__ __align__(16) float buf[2][H1_CHUNK];   /* double buffer: 32 KB */
  __shared__ u32 lh[2048];
  const int tid = threadIdx.x;
  for (int i = tid; i < 2048; i += 256) lh[i] = 0u;

  const int c0 = blockIdx.x * H1_CPB;
  const int b  = ((c0 * H1_CHUNK) / LOCS) & (BATCH - 1);   /* segment = a*4+b -> b */

  if ((tid >> 5) == 0)                       /* wave 0 drives the TDM */
    tdm_issue(obj + (u64)c0 * H1_CHUNK, buf[0]);
  __syncthreads();                           /* also covers lh init */

  for (int cc = 0; cc < H1_CPB; ++cc) {
    const int cur = cc & 1;
    if ((tid >> 5) == 0) {
      if (cc + 1 < H1_CPB) {
        tdm_issue(obj + (u64)(c0 + cc + 1) * H1_CHUNK, buf[cur ^ 1]);
        __builtin_amdgcn_s_wait_tensorcnt(1);   /* in-order: oldest (cur) has landed */
      } else {
        __builtin_amdgcn_s_wait_tensorcnt(0);
      }
    }
    __syncthreads();
    /* consume tile `cur` (ds_load_b128) while tile cur^1 streams in via TDM */
    const float4* src = (const float4*)buf[cur];
#pragma unroll
    for (int it = 0; it < H1_CHUNK / 4 / 256; ++it) {
      float4 v = src[tid + it * 256];
      atomicAdd(&lh[fkey(v.x) >> 21], 1u);
      atomicAdd(&lh[fkey(v.y) >> 21], 1u);
      atomicAdd(&lh[fkey(v.z) >> 21], 1u);
      atomicAdd(&lh[fkey(v.w) >> 21], 1u);
    }
    __syncthreads();
  }
  u32* gh = ws + OFF_HIST1 + b * 2048;
  for (int i = tid; i < 2048; i += 256) {
    u32 v = lh[i];
    if (v) atomicAdd(&gh[i], v);
  }
}

/* ================= radix-select passes 2/3 + selectors ================= */
__global__ void k_sel1(u32* ws) {
  int b = threadIdx.x; if (b >= BATCH) return;
  const u32* h = ws + OFF_HIST1 + b * 2048;
  u32 cum = 0;
  for (int bin = 2047; bin >= 0; --bin) {
    u32 c = h[bin];
    if (cum + c >= PRE_NMS) { ws[OFF_SEL+b*8+0] = (u32)bin; ws[OFF_SEL+b*8+1] = PRE_NMS - cum; return; }
    cum += c;
  }
}

__global__ __launch_bounds__(256) void k_hist2(const float* __restrict__ obj, u32* __restrict__ ws) {
  u32 stride = gridDim.x * blockDim.x;
  for (u32 f = blockIdx.x * blockDim.x + threadIdx.x; f < TOTAL; f += stride) {
    u32 r = f & (HT_TOT - 1);
    int b = (int)(r >> 18);
    u32 key = fkey(obj[f]);
    if ((key >> 21) == ws[OFF_SEL + b*8 + 0])
      atomicAdd(ws + OFF_HIST2 + b*2048 + ((key >> 10) & 0x7FFu), 1u);
  }
}

__global__ void k_sel2(u32* ws) {
  int b = threadIdx.x; if (b >= BATCH) return;
  const u32* h = ws + OFF_HIST2 + b * 2048;
  u32 need = ws[OFF_SEL + b*8 + 1];
  u32 cum = 0;
  for (int bin = 2047; bin >= 0; --bin) {
    u32 c = h[bin];
    if (cum + c >= need) { ws[OFF_SEL+b*8+2] = (u32)bin; ws[OFF_SEL+b*8+3] = need - cum; return; }
    cum += c;
  }
}

__global__ __launch_bounds__(256) void k_hist3(const float* __restrict__ obj, u32* __restrict__ ws) {
  u32 stride = gridDim.x * blockDim.x;
  for (u32 f = blockIdx.x * blockDim.x + threadIdx.x; f < TOTAL; f += stride) {
    u32 r = f & (HT_TOT - 1);
    int b = (int)(r >> 18);
    u32 key = fkey(obj[f]);
    u32 hi = (ws[OFF_SEL+b*8+0] << 11) | ws[OFF_SEL+b*8+2];
    if ((key >> 10) == hi)
      atomicAdd(ws + OFF_HIST3 + b*1024 + (key & 0x3FFu), 1u);
  }
}

__global__ void k_sel3(u32* ws) {
  int b = threadIdx.x; if (b >= BATCH) return;
  const u32* h = ws + OFF_HIST3 + b * 1024;
  u32 need = ws[OFF_SEL + b*8 + 3];
  u32 bin1 = ws[OFF_SEL + b*8 + 0], bin2 = ws[OFF_SEL + b*8 + 2];
  u32 cum = 0;
  for (int bin = 1023; bin >= 0; --bin) {
    u32 c = h[bin];
    if (cum + c >= need) {
      u32 eq  = need - cum;
      u32 kth = (bin1 << 21) | (bin2 << 10) | (u32)bin;
      ws[OFF_SEL+b*8+4] = kth;
      ws[OFF_SEL+b*8+5] = PRE_NMS - eq;   /* count strictly greater */
      ws[OFF_SEL+b*8+6] = eq;
      return;
    }
    cum += c;
  }
}

/* ================= compaction: exact top-2048 slots ================= */
__global__ __launch_bounds__(256) void k_compact(const float* __restrict__ obj, u32* __restrict__ ws) {
  u32 stride = gridDim.x * blockDim.x;
  for (u32 f = blockIdx.x * blockDim.x + threadIdx.x; f < TOTAL; f += stride) {
    u32 a   = f >> 20;             /* HT = 2^20 */
    u32 r   = f & (HT_TOT - 1);
    int b   = (int)(r >> 18);      /* LOCS = 2^18 */
    u32 loc = r & (LOCS - 1);
    u32 i   = loc * A_NUM + a;
    u32 key = fkey(obj[f]);
    u32 kth = ws[OFF_SEL + b*8 + 4];
    if (key > kth) {
      u32 s = atomicAdd(ws + OFF_CNT + b*2 + 0, 1u);
      ws[OFF_TKEY + b*PRE_NMS + s] = key;
      ws[OFF_TIDX + b*PRE_NMS + s] = i;
    } else if (key == kth) {
      u32 p = atomicAdd(ws + OFF_CNT + b*2 + 1, 1u);
      if (p < ws[OFF_SEL + b*8 + 6]) {
        u32 s = ws[OFF_SEL + b*8 + 5] + p;
        ws[OFF_TKEY + b*PRE_NMS + s] = key;
        ws[OFF_TIDX + b*PRE_NMS + s] = i;
      }
    }
  }
}

/* ================= LDS bitonic sort of 2048 (key desc, idx asc) ================= */
__global__ __launch_bounds__(1024) void k_sort(u32* __restrict__ ws) {
  __shared__ u64 arr[PRE_NMS];
  const int b = blockIdx.x, tid = threadIdx.x;
  for (int t = tid; t < PRE_NMS; t += 1024)
    arr[t] = ((u64)ws[OFF_TKEY + b*PRE_NMS + t] << 32) | (u32)(~ws[OFF_TIDX + b*PRE_NMS + t]);
  for (int k = 2; k <= PRE_NMS; k <<= 1) {
    for (int j = k >> 1; j > 0; j >>= 1) {
      __syncthreads();
      for (int t = tid; t < PRE_NMS; t += 1024) {
        int x = t ^ j;
        if (x > t) {
          u64 a0 = arr[t], a1 = arr[x];
          bool desc = (t & k) == 0;
          if (desc ? (a0 < a1) : (a0 > a1)) { arr[t] = a1; arr[x] = a0; }
        }
      }
    }
  }
  __syncthreads();
  float* sc = (float*)ws + OFF_SCORE + b * PRE_NMS;
  for (int t = tid; t < PRE_NMS; t += 1024) {
    u64 v = arr[t];
    u32 key = (u32)(v >> 32), idx = ~(u32)v;
    ws[OFF_TIDX + b*PRE_NMS + t] = idx;
    float lg = kinv(key);
    sc[t] = 1.0f / (1.0f + expf(-lg));   /* sigmoid */
  }
}

/* ================= gather + decode + AABB/vol precompute ================= */
__global__ __launch_bounds__(256) void k_decode(const float* __restrict__ breg,
                                                const float* __restrict__ anchors,
                                                u32* __restrict__ ws) {
  int t = blockIdx.x * blockDim.x + threadIdx.x;
  if (t >= BATCH * PRE_NMS) return;
  int b = t >> 11, k = t & (PRE_NMS - 1);
  u32 i  = ws[OFF_TIDX + b*PRE_NMS + k];
  u32 a  = i & (A_NUM - 1);
  u32 lc = i >> 1;
  u64 row = (u64)b * LOCS + lc;
  float d[7];
#pragma unroll
  for (int c = 0; c < 7; ++c) d[c] = breg[(u64)(a*7 + c) * HT_TOT + row];
  const float* an = anchors + ((u64)b * NPB + i) * 7;
  float xa = an[0], ya = an[1], za = an[2];
  float wa = an[3], la = an[4], ha = an[5], ra = an[6];
  float diag = sqrtf(wa*wa + la*la);
  float x = d[0]*diag + xa;
  float y = d[1]*diag + ya;
  float z = d[2]*ha   + za;
  float w = expf(d[3]) * wa;
  float l = expf(d[4]) * la;
  float h = expf(d[5]) * ha;
  float rr = d[6] + ra;
  float* bx = (float*)ws + OFF_BOX + (size_t)(b*PRE_NMS + k) * 7;
  bx[0]=x; bx[1]=y; bx[2]=z; bx[3]=w; bx[4]=l; bx[5]=h; bx[6]=rr;
  float* ax = (float*)ws + OFF_AUX + (size_t)(b*PRE_NMS + k) * 7;
  ax[0]=x - 0.5f*w; ax[1]=x + 0.5f*w;
  ax[2]=y - 0.5f*l; ax[3]=y + 0.5f*l;
  ax[4]=z;          ax[5]=z + h;
  ax[6]=fmaxf(w,0.f)*fmaxf(l,0.f)*fmaxf(h,0.f);
}

/* ================= suppression matrix: one wave = one (row, 32-col word) ================= */
__global__ __launch_bounds__(256) void k_sup(u32* __restrict__ ws) {
  const float* aux = (const float*)ws + OFF_AUX;
  int wave = blockIdx.x * (blockDim.x >> 5) + (threadIdx.x >> 5);
  int lane = threadIdx.x & 31;
  int b   = wave >> 17;             /* 2048*64 = 2^17 waves per batch */
  int rem = wave & 131071;
  int i   = rem >> 6;
  int wj  = rem & 63;
  int j   = wj * 32 + lane;
  const float* ai = aux + (size_t)(b*PRE_NMS + i) * 7;
  const float* aj = aux + (size_t)(b*PRE_NMS + j) * 7;
  float ox = fmaxf(fminf(ai[1], aj[1]) - fmaxf(ai[0], aj[0]), 0.f);
  float oy = fmaxf(fminf(ai[3], aj[3]) - fmaxf(ai[2], aj[2]), 0.f);
  float oz = fmaxf(fminf(ai[5], aj[5]) - fmaxf(ai[4], aj[4]), 0.f);
  float inter = ox * oy * oz;
  float iou = inter / (ai[6] + aj[6] - inter + 1e-8f);
  bool cond = (j > i) && (iou > NMS_THR);
#if __has_builtin(__builtin_amdgcn_ballot_w32)
  u32 m = __builtin_amdgcn_ballot_w32(cond);
#else
  u32 m = (u32)__ballot(cond);
#endif
  if (lane == 0) ws[OFF_SUP + (size_t)(b*PRE_NMS + i) * 64 + wj] = m;
}

/* ================= greedy bitmask scan + rank-prefix output ================= */
__global__ __launch_bounds__(64) void k_scan(u32* __restrict__ ws, float* __restrict__ out) {
  __shared__ u32 keepw[64];
  __shared__ u32 kpref[64];
  __shared__ u32 total;
  const int b = blockIdx.x, tid = threadIdx.x;
  keepw[tid] = 0xFFFFFFFFu;
  const u32* sup = ws + OFF_SUP + (size_t)b * PRE_NMS * 64;
  for (int i = 0; i < PRE_NMS; ++i) {
    __syncthreads();
    u32 kw = keepw[i >> 5];
    if ((kw >> (i & 31)) & 1u)
      keepw[tid] &= ~sup[(size_t)i * 64 + tid];   /* row i has bits only for j>i */
  }
  __syncthreads();
  if (tid == 0) {
    u32 c = 0;
    for (int w = 0; w < 64; ++w) { kpref[w] = c; c += __popc(keepw[w]); }
    total = c;
  }
  __syncthreads();
  const float* boxes = (const float*)ws + OFF_BOX + (size_t)b * PRE_NMS * 7;
  const float* sc    = (const float*)ws + OFF_SCORE + (size_t)b * PRE_NMS;
  for (int k = tid; k < PRE_NMS; k += 64) {
    int w = k >> 5;
    u32 kept = (keepw[w] >> (k & 31)) & 1u;
    u32 keptBefore = kpref[w] + __popc(keepw[w] & ((1u << (k & 31)) - 1u));
    int slot; float s;
    if (kept) { slot = (int)keptBefore; s = sc[k]; }
    else      { slot = (int)total + (k - (int)keptBefore); s = 0.0f; }  /* -inf -> 0 fill */
    if (slot < POST_NMS) {
      float* o = out + ((size_t)b * POST_NMS + slot) * 8;
#pragma unroll
      for (int c = 0; c < 7; ++c) o[c] = boxes[(size_t)k * 7 + c];
      o[7] = s;
    }
  }
}

extern "C" void kernel_launch(void* const* d_in, const int* in_sizes, int n_in,
                              void* d_out, int out_size, void* d_ws, size_t ws_size,
                              hipStream_t stream) {
  const float* obj     = (const float*)d_in[0];
  const float* breg    = (const float*)d_in[1];
  const float* anchors = (const float*)d_in[2];
  float* out = (float*)d_out;
  u32* ws = (u32*)d_ws;
  (void)in_sizes; (void)n_in; (void)out_size; (void)ws_size;

  hipMemsetAsync(d_ws, 0, (size_t)ZERO_U32 * sizeof(u32), stream);
  k_hist1  <<<TOTAL / (H1_CHUNK * H1_CPB), 256, 0, stream>>>(obj, ws);
  k_sel1   <<<1, BATCH, 0, stream>>>(ws);
  k_hist2  <<<512, 256, 0, stream>>>(obj, ws);
  k_sel2   <<<1, BATCH, 0, stream>>>(ws);
  k_hist3  <<<512, 256, 0, stream>>>(obj, ws);
  k_sel3   <<<1, BATCH, 0, stream>>>(ws);
  k_compact<<<512, 256, 0, stream>>>(obj, ws);
  k_sort   <<<BATCH, 1024, 0, stream>>>(ws);
  k_decode <<<(BATCH * PRE_NMS + 255) / 256, 256, 0, stream>>>(breg, anchors, ws);
  k_sup    <<<BATCH * PRE_NMS * 64 / 8, 256, 0, stream>>>(ws);
  k_scan   <<<BATCH, 64, 0, stream>>>(ws, out);
}